// AbstractAffine_79611513799169
// MI455X (gfx1250) — compile-verified
//
#include <hip/hip_runtime.h>

#define D 1024
#define L 16
#define BM 64
#define BN 64
#define BK 32
#define DD (D * D)

typedef float v2f __attribute__((ext_vector_type(2)));
typedef float v8f __attribute__((ext_vector_type(8)));

// Ping-pong scratch: [ping][lower/upper]. Static device globals so we do not
// depend on ws_size; written before read every call (deterministic).
__device__ float g_x[2][2][DD];
__device__ float g_b[2][2][D];

// ---------------------------------------------------------------------------
// Async Global->LDS copy of 16 bytes (GLOBAL_LOAD_ASYNC_TO_LDS_B128).
// VDST = LDS byte address (addr[31:0] of the flat LDS pointer), VADDR = 64-bit
// global address. Tracked by ASYNCcnt; completed via s_wait_asynccnt.
// ---------------------------------------------------------------------------
__device__ __forceinline__ void async_copy_b128(void* lds, const void* gptr) {
  const unsigned int loff = (unsigned int)(unsigned long long)lds;  // LDS offset
  asm volatile("global_load_async_to_lds_b128 %0, %1, off"
               :
               : "v"(loff), "v"(gptr)
               : "memory");
}

__device__ __forceinline__ void wait_async_all() {
  asm volatile("s_wait_asynccnt 0x0" ::: "memory");
}

// ---------------------------------------------------------------------------
// x0 = W^T (both lower and upper start identical); b0 = biases.
// ---------------------------------------------------------------------------
__global__ __launch_bounds__(256) void k_init(const float* __restrict__ W,
                                              const float* __restrict__ biases) {
  __shared__ float tile[32][33];
  const int bx = blockIdx.x * 32;  // output row block (m)
  const int by = blockIdx.y * 32;  // output col block (k)
  const int tx = threadIdx.x & 31;
  const int ty = threadIdx.x >> 5;  // 0..7
  for (int i = 0; i < 32; i += 8)
    tile[ty + i][tx] = W[(size_t)(by + ty + i) * D + bx + tx];
  __syncthreads();
  for (int i = 0; i < 32; i += 8) {
    const float v = tile[tx][ty + i];  // = W[by+tx][bx+ty+i]
    const size_t idx = (size_t)(bx + ty + i) * D + (by + tx);
    g_x[0][0][idx] = v;
    g_x[0][1][idx] = v;
  }
  if (blockIdx.x == 0 && blockIdx.y == 0) {
    for (int i = threadIdx.x; i < D; i += 256) {
      const float b = biases[i];
      g_b[0][0][i] = b;
      g_b[0][1][i] = b;
    }
  }
}

// ---------------------------------------------------------------------------
// One backsubstitution layer:
//   Yl = max(Xl,0) @ Alo + min(Xl,0) @ Aup
//   Yu = max(Xu,0) @ Aup + min(Xu,0) @ Alo
// fp32 WMMA (V_WMMA_F32_16X16X4_F32). 128 threads = 4 waves; block tile 64x64;
// each wave computes a 32x32 tile of BOTH Yl and Yu (shares Alo/Aup fragments).
// Double-buffered LDS filled by async Global->LDS DMA, pipelined one K-stage
// ahead of the WMMA loop.
// ---------------------------------------------------------------------------
__global__ __launch_bounds__(128) void k_gemm(const float* __restrict__ Alo,
                                              const float* __restrict__ Aup,
                                              int src) {
  const float* __restrict__ Xl = g_x[src][0];
  const float* __restrict__ Xu = g_x[src][1];
  float* __restrict__ Yl = g_x[src ^ 1][0];
  float* __restrict__ Yu = g_x[src ^ 1][1];

  __shared__ float sXl[2][BM][BK];
  __shared__ float sXu[2][BM][BK];
  __shared__ float sAlo[2][BK][BN];
  __shared__ float sAup[2][BK][BN];

  const int tid = threadIdx.x;
  const int lane = tid & 31;
  const int wave = tid >> 5;           // 0..3
  const int wm = (wave & 1) * 32;      // wave row offset in block tile
  const int wn = (wave >> 1) * 32;     // wave col offset in block tile
  const int bm = blockIdx.y * BM;
  const int bn = blockIdx.x * BN;
  const int lm = lane & 15;            // matrix row/col within 16-tile
  const int lk2 = (lane >> 4) * 2;     // K base from lane half (A & B layout)

  // Per-thread staging coordinates (128 threads move 512 float4 per tile).
  const int xr = tid >> 3;             // 0..15  (X tiles: row step 16)
  const int xc = (tid & 7) * 4;        // 0..28
  const int ar = tid >> 4;             // 0..7   (A tiles: row step 8)
  const int ac = (tid & 15) * 4;       // 0..60

  v8f accl[2][2] = {};
  v8f accu[2][2] = {};

  // Issue async stage for K-chunk k0 into LDS buffer `buf`.
  auto issue_stage = [&](int buf, int k0) {
#pragma unroll
    for (int it = 0; it < 4; ++it) {
      const int r = xr + it * 16;
      async_copy_b128(&sXl[buf][r][xc], &Xl[(size_t)(bm + r) * D + k0 + xc]);
      async_copy_b128(&sXu[buf][r][xc], &Xu[(size_t)(bm + r) * D + k0 + xc]);
    }
#pragma unroll
    for (int it = 0; it < 4; ++it) {
      const int r = ar + it * 8;
      async_copy_b128(&sAlo[buf][r][ac], &Alo[(size_t)(k0 + r) * D + bn + ac]);
      async_copy_b128(&sAup[buf][r][ac], &Aup[(size_t)(k0 + r) * D + bn + ac]);
    }
  };

  issue_stage(0, 0);
  int cur = 0;

  for (int k0 = 0; k0 < D; k0 += BK) {
    // Publish this wave's async fills, then make all waves' fills visible and
    // guarantee nobody still reads the buffer we are about to prefetch into.
    wait_async_all();
    __syncthreads();
    if (k0 + BK < D) issue_stage(cur ^ 1, k0 + BK);

#pragma unroll
    for (int kk = 0; kk < BK; kk += 4) {
      // B fragments: b[v] = B[kk + lk2 + v][n]
      v2f fblo[2], fbup[2];
#pragma unroll
      for (int ct = 0; ct < 2; ++ct) {
        const int n = wn + ct * 16 + lm;
        v2f t0 = {sAlo[cur][kk + lk2 + 0][n], sAlo[cur][kk + lk2 + 1][n]};
        v2f t1 = {sAup[cur][kk + lk2 + 0][n], sAup[cur][kk + lk2 + 1][n]};
        fblo[ct] = t0;
        fbup[ct] = t1;
      }
      // A fragments with pos/neg split: a[v] = X[m][kk + lk2 + v]
      v2f fpl[2], fnl[2], fpu[2], fnu[2];
#pragma unroll
      for (int rt = 0; rt < 2; ++rt) {
        const int m = wm + rt * 16 + lm;
        const float2 xlv = *(const float2*)&sXl[cur][m][kk + lk2];
        const float2 xuv = *(const float2*)&sXu[cur][m][kk + lk2];
        v2f pl = {fmaxf(xlv.x, 0.f), fmaxf(xlv.y, 0.f)};
        v2f nl = {fminf(xlv.x, 0.f), fminf(xlv.y, 0.f)};
        v2f pu = {fmaxf(xuv.x, 0.f), fmaxf(xuv.y, 0.f)};
        v2f nu = {fminf(xuv.x, 0.f), fminf(xuv.y, 0.f)};
        fpl[rt] = pl; fnl[rt] = nl; fpu[rt] = pu; fnu[rt] = nu;
      }
#pragma unroll
      for (int rt = 0; rt < 2; ++rt) {
#pragma unroll
        for (int ct = 0; ct < 2; ++ct) {
          accl[rt][ct] = __builtin_amdgcn_wmma_f32_16x16x4_f32(
              false, fpl[rt], false, fblo[ct], (short)0, accl[rt][ct], false, false);
          accl[rt][ct] = __builtin_amdgcn_wmma_f32_16x16x4_f32(
              false, fnl[rt], false, fbup[ct], (short)0, accl[rt][ct], false, false);
          accu[rt][ct] = __builtin_amdgcn_wmma_f32_16x16x4_f32(
              false, fpu[rt], false, fbup[ct], (short)0, accu[rt][ct], false, false);
          accu[rt][ct] = __builtin_amdgcn_wmma_f32_16x16x4_f32(
              false, fnu[rt], false, fblo[ct], (short)0, accu[rt][ct], false, false);
        }
      }
    }
    cur ^= 1;
  }

  // Epilogue: C/D layout -> m = base + e + 8*(lane>=16), n = base + (lane&15).
  const int mhalf = (lane >> 4) * 8;
#pragma unroll
  for (int rt = 0; rt < 2; ++rt) {
#pragma unroll
    for (int ct = 0; ct < 2; ++ct) {
      const int n = bn + wn + ct * 16 + lm;
#pragma unroll
      for (int e = 0; e < 8; ++e) {
        const int m = bm + wm + rt * 16 + mhalf + e;
        Yl[(size_t)m * D + n] = accl[rt][ct][e];
        Yu[(size_t)m * D + n] = accu[rt][ct][e];
      }
    }
  }
}

// ---------------------------------------------------------------------------
// Intercept accumulation: bl += pl@c_lo + nl@c_up ; bu += pu@c_up + nu@c_lo.
// One wave per output row; coalesced along k.
// ---------------------------------------------------------------------------
__global__ __launch_bounds__(256) void k_bias(const float* __restrict__ clo,
                                              const float* __restrict__ cup,
                                              int src) {
  const int row = blockIdx.x * 8 + (threadIdx.x >> 5);
  const int lane = threadIdx.x & 31;
  const float* __restrict__ Xl = g_x[src][0];
  const float* __restrict__ Xu = g_x[src][1];
  float sl = 0.f, su = 0.f;
  for (int k = lane; k < D; k += 32) {
    const float xl = Xl[(size_t)row * D + k];
    const float xu = Xu[(size_t)row * D + k];
    const float cl = clo[k];
    const float cu = cup[k];
    sl += fmaxf(xl, 0.f) * cl + fminf(xl, 0.f) * cu;
    su += fmaxf(xu, 0.f) * cu + fminf(xu, 0.f) * cl;
  }
  for (int off = 16; off > 0; off >>= 1) {
    sl += __shfl_down(sl, off, 32);
    su += __shfl_down(su, off, 32);
  }
  if (lane == 0) {
    g_b[src ^ 1][0][row] = g_b[src][0][row] + sl;
    g_b[src ^ 1][1][row] = g_b[src][1][row] + su;
  }
}

// ---------------------------------------------------------------------------
// Concretize against the input box. out[0:D] = lowers, out[D:2D] = uppers.
// ---------------------------------------------------------------------------
__global__ __launch_bounds__(256) void k_final(const float* __restrict__ ilo,
                                               const float* __restrict__ iup,
                                               float* __restrict__ out) {
  const int row = blockIdx.x * 8 + (threadIdx.x >> 5);
  const int lane = threadIdx.x & 31;
  const float* __restrict__ Xl = g_x[0][0];
  const float* __restrict__ Xu = g_x[0][1];
  float sl = 0.f, su = 0.f;
  for (int n = lane; n < D; n += 32) {
    const float xl = Xl[(size_t)row * D + n];
    const float xu = Xu[(size_t)row * D + n];
    const float lo = ilo[n];
    const float up = iup[n];
    sl += fmaxf(xl, 0.f) * lo + fminf(xl, 0.f) * up;
    su += fmaxf(xu, 0.f) * up + fminf(xu, 0.f) * lo;
  }
  for (int off = 16; off > 0; off >>= 1) {
    sl += __shfl_down(sl, off, 32);
    su += __shfl_down(su, off, 32);
  }
  if (lane == 0) {
    out[row] = sl + g_b[0][0][row];
    out[D + row] = su + g_b[0][1][row];
  }
}

extern "C" void kernel_launch(void* const* d_in, const int* in_sizes, int n_in,
                              void* d_out, int out_size, void* d_ws, size_t ws_size,
                              hipStream_t stream) {
  (void)in_sizes; (void)n_in; (void)out_size; (void)d_ws; (void)ws_size;
  const float* weights = (const float*)d_in[0];
  const float* biases  = (const float*)d_in[1];
  const float* nxl     = (const float*)d_in[2];
  const float* nxu     = (const float*)d_in[3];
  const float* nbl     = (const float*)d_in[4];
  const float* nbu     = (const float*)d_in[5];
  const float* ilo     = (const float*)d_in[6];
  const float* iup     = (const float*)d_in[7];
  float* out = (float*)d_out;

  k_init<<<dim3(32, 32), 256, 0, stream>>>(weights, biases);

  int ping = 0;
  for (int i = 0; i < L; ++i) {
    k_gemm<<<dim3(D / BN, D / BM), 128, 0, stream>>>(
        nxl + (size_t)i * DD, nxu + (size_t)i * DD, ping);
    k_bias<<<dim3(D / 8), 256, 0, stream>>>(
        nbl + (size_t)i * D, nbu + (size_t)i * D, ping);
    ping ^= 1;
  }

  k_final<<<dim3(D / 8), 256, 0, stream>>>(ilo, iup, out);
}